// MTAAttentionTriton_79121887527345
// MI455X (gfx1250) — compile-verified
//
#include <hip/hip_runtime.h>
#include <cstdint>
#include <cstddef>

#define HH 16
#define SS 2048
#define DD 64

typedef __attribute__((ext_vector_type(2))) float v2f;
typedef __attribute__((ext_vector_type(4))) float v4f;
typedef __attribute__((ext_vector_type(8))) float v8f;

#if __has_builtin(__builtin_amdgcn_global_load_async_to_lds_b32) && \
    __has_builtin(__builtin_amdgcn_s_wait_asynccnt)
#define USE_ASYNC_LDS 1
#else
#define USE_ASYNC_LDS 0
#endif

// ---------------------------------------------------------------------------
// Kernel 1: S0[h,q,k] = (1/8) * dot(Q[h,q,:], K[h,k,:]),  zeroed where k > q
// One wave (32 lanes) per 16x16 tile, V_WMMA_F32_16X16X4_F32 over D=64.
// ---------------------------------------------------------------------------
__global__ __launch_bounds__(256) void qk_scores_kernel(
    const float* __restrict__ Q, const float* __restrict__ K,
    float* __restrict__ S0)
{
  const int wave = threadIdx.x >> 5;
  const int lane = threadIdx.x & 31;
  const int m    = lane & 15;
  const int half = lane >> 4;
  const int ktile = blockIdx.x * 8 + wave;
  const int qtile = blockIdx.y;
  const int h     = blockIdx.z;
  const int qb = qtile * 16, kb = ktile * 16;

  float* out = S0 + ((size_t)h * SS + qb) * SS + kb;

  if (kb > qb + 15) {  // fully non-causal tile: conv input must read zeros
#pragma unroll
    for (int r = 0; r < 8; ++r)
      out[(size_t)(r + half * 8) * SS + m] = 0.f;
    return;
  }

  // A = Q tile (M=q, K=d); lanes 0-15: d = d0+{0,1}; lanes 16-31: d = d0+{2,3}
  const v2f* qrow = (const v2f*)(Q + ((size_t)h * SS + qb + m) * DD);
  // B = K^T tile (K=d, N=k); lane picks N=k-row of K, VGPR picks d
  const v2f* krow = (const v2f*)(K + ((size_t)h * SS + kb + m) * DD);

  v8f acc = {};
#pragma unroll
  for (int d0 = 0; d0 < DD; d0 += 4) {
    v2f a = qrow[(d0 >> 1) + half];
    v2f b = krow[(d0 >> 1) + half];
    acc = __builtin_amdgcn_wmma_f32_16x16x4_f32(
        false, a, false, b, (short)0, acc, false, false);
  }

#pragma unroll
  for (int r = 0; r < 8; ++r) {
    int q = qb + r + half * 8;
    int k = kb + m;
    float v = acc[r] * 0.125f;   // 1/sqrt(64)
    if (k > q) v = 0.f;          // jnp.where(causal, scores, 0) pre-conv
    out[(size_t)(r + half * 8) * SS + m] = v;
  }
}

// ---------------------------------------------------------------------------
// Kernels 2 & 4: fused depthwise causal conv (KQ x KK) + per-head mixing.
// One block handles a 16q x 16k tile for ALL 16 heads (mixing needs all heads
// at the same (q,k)). Input tile + halo staged in LDS — via CDNA5
// GLOBAL_LOAD_ASYNC_TO_LDS_B32 when the builtin is available.
// NEGINF: pre-softmax path masks non-causal to -inf; post path masks to 0.
// ---------------------------------------------------------------------------
template <int KQ, int KK, bool NEGINF>
__global__ __launch_bounds__(256) void conv_mix_kernel(
    const float* __restrict__ in, const float* __restrict__ wconv,
    const float* __restrict__ wmix, float* __restrict__ out)
{
  constexpr int QT = 16, KT = 16;
  constexpr int QI = QT + KQ - 1;
  constexpr int KI = KT + KK - 1;

  __shared__ float sIn[HH][QI][KI];   // input tile + causal halo (top/left)
  __shared__ float sMid[HH][QT][KT];  // depthwise conv result per head
  __shared__ float sW[HH][KQ][KK];    // conv weights
  __shared__ float sM[HH][HH];        // head mixing matrix

  const int tid = threadIdx.x;
  const int kb = blockIdx.x * KT;
  const int qb = blockIdx.y * QT;
  const float maskv = NEGINF ? -__builtin_inff() : 0.f;

  if (kb > qb + QT - 1) {  // fully non-causal tile: just write mask value
    for (int i = tid; i < HH * QT * KT; i += 256) {
      int g = i >> 8, rem = i & 255, qq = rem >> 4, kk = rem & 15;
      out[((size_t)g * SS + qb + qq) * SS + kb + kk] = maskv;
    }
    return;
  }

  for (int i = tid; i < HH * KQ * KK; i += 256) (&sW[0][0][0])[i] = wconv[i];
  for (int i = tid; i < HH * HH; i += 256)      (&sM[0][0])[i]    = wmix[i];

  // Stage input tile + halo into LDS. In-bounds elements go through the
  // async DMA path (no VGPR round-trip, ASYNCcnt-tracked); out-of-bounds
  // halo is zero-filled with plain LDS stores (disjoint addresses).
  for (int i = tid; i < HH * QI * KI; i += 256) {
    int h = i / (QI * KI), rem = i % (QI * KI);
    int qq = rem / KI, kk = rem % KI;
    int q = qb - (KQ - 1) + qq;
    int k = kb - (KK - 1) + kk;
    float* ldst = &sIn[h][qq][kk];
    if (q >= 0 && k >= 0) {
      const float* gsrc = in + ((size_t)h * SS + q) * SS + k;
#if USE_ASYNC_LDS
      __builtin_amdgcn_global_load_async_to_lds_b32(
          (__attribute__((address_space(1))) int*)gsrc,
          (__attribute__((address_space(3))) int*)ldst, 0, 0);
#else
      *ldst = *gsrc;
#endif
    } else {
      *ldst = 0.f;
    }
  }
#if USE_ASYNC_LDS
  __builtin_amdgcn_s_wait_asynccnt(0);
#endif
  __syncthreads();

  // Depthwise causal conv: out(q,k) = sum_ab w[a][b] * in(q-(KQ-1)+a, k-(KK-1)+b)
  for (int i = tid; i < HH * QT * KT; i += 256) {
    int h = i >> 8, rem = i & 255, qq = rem >> 4, kk = rem & 15;
    float acc = 0.f;
#pragma unroll
    for (int a = 0; a < KQ; ++a)
#pragma unroll
      for (int b = 0; b < KK; ++b)
        acc = fmaf(sW[h][a][b], sIn[h][qq + a][kk + b], acc);
    sMid[h][qq][kk] = acc;
  }
  __syncthreads();

  // Per-head linear mixing: out[g] = sum_h wmix[g][h] * conv[h], then mask
  for (int i = tid; i < HH * QT * KT; i += 256) {
    int g = i >> 8, rem = i & 255, qq = rem >> 4, kk = rem & 15;
    int q = qb + qq, k = kb + kk;
    float acc = 0.f;
#pragma unroll
    for (int h = 0; h < HH; ++h)
      acc = fmaf(sM[g][h], sMid[h][qq][kk], acc);
    if (k > q) acc = maskv;
    out[((size_t)g * SS + q) * SS + k] = acc;
  }
}

// ---------------------------------------------------------------------------
// Kernel 3: in-place row softmax over k. One wave per (h,q) row of 2048.
// Vectorized float4 accesses: each lane owns contiguous 16B chunks, the wave
// covers full cachelines. Non-causal entries are -inf -> exactly 0 on output.
// ---------------------------------------------------------------------------
__global__ __launch_bounds__(256) void softmax_kernel(float* __restrict__ buf)
{
  const int wave = threadIdx.x >> 5;
  const int lane = threadIdx.x & 31;
  const int row  = blockIdx.x * 8 + wave;
  v4f* r4 = (v4f*)(buf + (size_t)row * SS);
  constexpr int NV = SS / 4;  // 512 float4 per row

  float mx = -__builtin_inff();
  for (int j = lane; j < NV; j += 32) {
    v4f v = r4[j];
    mx = fmaxf(mx, fmaxf(fmaxf(v.x, v.y), fmaxf(v.z, v.w)));
  }
#pragma unroll
  for (int off = 16; off > 0; off >>= 1)
    mx = fmaxf(mx, __shfl_xor(mx, off, 32));

  float sum = 0.f;
  for (int j = lane; j < NV; j += 32) {
    v4f v = r4[j];
    v.x = __expf(v.x - mx);
    v.y = __expf(v.y - mx);
    v.z = __expf(v.z - mx);
    v.w = __expf(v.w - mx);
    r4[j] = v;
    sum += (v.x + v.y) + (v.z + v.w);
  }
#pragma unroll
  for (int off = 16; off > 0; off >>= 1)
    sum += __shfl_xor(sum, off, 32);

  float inv = 1.0f / sum;
  for (int j = lane; j < NV; j += 32) {
    v4f v = r4[j];
    v.x *= inv; v.y *= inv; v.z *= inv; v.w *= inv;
    r4[j] = v;
  }
}

// ---------------------------------------------------------------------------
// Kernel 5: O[q,h,d] = sum_k P[h,q,k] * V[h,k,d]. One wave per (h, q-tile),
// 4 WMMA accumulators cover D=64. k-loop stops at the diagonal (P==0 beyond).
// ---------------------------------------------------------------------------
__global__ __launch_bounds__(32) void pv_kernel(
    const float* __restrict__ P, const float* __restrict__ V,
    float* __restrict__ O)
{
  const int lane = threadIdx.x & 31;
  const int m    = lane & 15;
  const int half = lane >> 4;
  const int qb = blockIdx.x * 16;
  const int h  = blockIdx.y;

  const v2f*  prow = (const v2f*)(P + ((size_t)h * SS + qb + m) * SS);
  const float* vb  = V + (size_t)h * SS * DD;

  v8f acc0 = {}, acc1 = {}, acc2 = {}, acc3 = {};
  const int kmax = qb + 16;  // causal: P[q][k]==0 for k>q

  for (int k0 = 0; k0 < kmax; k0 += 4) {
    v2f a = prow[(k0 >> 1) + half];  // A: P tile (M=q, K=k)
    const float* v0 = vb + (size_t)(k0 + 2 * half) * DD + m;  // B: V (K=k, N=d)
    v2f b;
    b.x = v0[0];      b.y = v0[DD];
    acc0 = __builtin_amdgcn_wmma_f32_16x16x4_f32(false, a, false, b, (short)0, acc0, false, false);
    b.x = v0[16];     b.y = v0[DD + 16];
    acc1 = __builtin_amdgcn_wmma_f32_16x16x4_f32(false, a, false, b, (short)0, acc1, false, false);
    b.x = v0[32];     b.y = v0[DD + 32];
    acc2 = __builtin_amdgcn_wmma_f32_16x16x4_f32(false, a, false, b, (short)0, acc2, false, false);
    b.x = v0[48];     b.y = v0[DD + 48];
    acc3 = __builtin_amdgcn_wmma_f32_16x16x4_f32(false, a, false, b, (short)0, acc3, false, false);
  }

#pragma unroll
  for (int r = 0; r < 8; ++r) {
    int q = qb + r + half * 8;
    float* orow = O + ((size_t)q * HH + h) * DD + m;  // out layout [S,H,D]
    orow[0]  = acc0[r];
    orow[16] = acc1[r];
    orow[32] = acc2[r];
    orow[48] = acc3[r];
  }
}

// ---------------------------------------------------------------------------
// Launcher. Inputs (setup_inputs order):
//   0 xq [1,16,2048,64] f32      4 w_conv_pre [16,6,11] f32
//   1 xk [1,16,2048,64] f32      5 w_conv_post [16,3,3] f32
//   2 xv [1,16,2048,64] f32      6 w_psm  [16,16] f32
//   3 mask (unused)              7 w_posm [16,16] f32
// d_out: [1,2048,16,64] f32. Workspace: 2 x 256MB fp32 score buffers.
// ---------------------------------------------------------------------------
extern "C" void kernel_launch(void* const* d_in, const int* in_sizes, int n_in,
                              void* d_out, int out_size, void* d_ws, size_t ws_size,
                              hipStream_t stream)
{
  (void)in_sizes; (void)n_in; (void)out_size; (void)ws_size;
  const float* xq     = (const float*)d_in[0];
  const float* xk     = (const float*)d_in[1];
  const float* xv     = (const float*)d_in[2];
  const float* w_pre  = (const float*)d_in[4];
  const float* w_post = (const float*)d_in[5];
  const float* w_psm  = (const float*)d_in[6];
  const float* w_posm = (const float*)d_in[7];

  float* bufA = (float*)d_ws;                      // [H,S,S] fp32
  float* bufB = bufA + (size_t)HH * SS * SS;       // [H,S,S] fp32

  // 1) scores = scale * QK^T (causal-zeroed) -> A
  qk_scores_kernel<<<dim3(SS / 16 / 8, SS / 16, HH), 256, 0, stream>>>(xq, xk, bufA);
  // 2) pre-softmax conv(6x11) + head mix + (-inf mask): A -> B
  conv_mix_kernel<6, 11, true><<<dim3(SS / 16, SS / 16), 256, 0, stream>>>(bufA, w_pre, w_psm, bufB);
  // 3) softmax in-place on B
  softmax_kernel<<<dim3(HH * SS / 8), 256, 0, stream>>>(bufB);
  // 4) post-softmax conv(3x3) + head mix + (0 mask): B -> A
  conv_mix_kernel<3, 3, false><<<dim3(SS / 16, SS / 16), 256, 0, stream>>>(bufB, w_post, w_posm, bufA);
  // 5) O = P @ V, output [S,H,D]
  pv_kernel<<<dim3(SS / 16, HH), 32, 0, stream>>>(bufA, xv, (float*)d_out);
}